// GaussianScene_43542378447305
// MI455X (gfx1250) — compile-verified
//
#include <hip/hip_runtime.h>
#include <hip/hip_bf16.h>

typedef __attribute__((ext_vector_type(2))) float v2f;
typedef __attribute__((ext_vector_type(8))) float v8f;

// ---------------------------------------------------------------------------
// Kernel 0: initialize per-pixel winner buffer to -1 (no point yet).
// ---------------------------------------------------------------------------
__global__ void gs_fill_winner(int* __restrict__ winner, int npix) {
    int i = blockIdx.x * blockDim.x + threadIdx.x;
    if (i < npix) winner[i] = -1;
}

// ---------------------------------------------------------------------------
// Kernel 1: project points via V_WMMA_F32_16X16X4_F32 and atomicMax the point
// index into the winner buffer (== jax.ops.segment_max over arange(n)).
//
// WMMA mapping (D = A x B, 16x16 = 16x4 * 4x16):
//   A (16x4)  = camera transform: rows 0..2 = [R | t], rows 3..15 = 0.
//               Lane L holds row M=L%16; v0 = A[M][L<16?0:2], v1 = A[M][L<16?1:3].
//   B (4x16)  = 16 homogeneous points as columns: B[k][n] = hom coord k of
//               point n. Lanes 0..15 hold K=0,1 (x,y); lanes 16..31 K=2,3 (z,1).
//   D[m][n]   = transformed coord m of point n -> lanes 0..15 read X',Y',Z'
//               from d[0],d[1],d[2] for point n == lane.
// ---------------------------------------------------------------------------
__global__ void gs_project_wmma(const float* __restrict__ pos,   // n x 3
                                const float* __restrict__ cam,   // 4 x 4
                                const float* __restrict__ intr,  // 3 x 3
                                const int*   __restrict__ Hp,
                                const int*   __restrict__ Wp,
                                int* __restrict__ winner,
                                int n) {
    const int H  = *Hp;
    const int W  = *Wp;
    const float fx = intr[0], cx = intr[2];
    const float fy = intr[4], cy = intr[5];

    const int lane = threadIdx.x & 31;
    const int m    = lane & 15;          // matrix row (A) / point slot (B)
    const bool top = lane < 16;

    // A-matrix: uniform per wave, built once. Row m<3: [R[m][0..2], t[m]].
    v2f a;
    a.x = (m < 3) ? cam[m * 4 + (top ? 0 : 2)] : 0.0f;   // K = 0 or 2
    a.y = (m < 3) ? cam[m * 4 + (top ? 1 : 3)] : 0.0f;   // K = 1 or 3 (t in K=3)

    const int waveId = (blockIdx.x * blockDim.x + threadIdx.x) >> 5;
    const int nWaves = (gridDim.x * blockDim.x) >> 5;

    for (long long base = (long long)waveId * 16; base < n;
         base += (long long)nWaves * 16) {
        const int  pidx = (int)base + m;
        const bool inR  = pidx < n;

        float px = 0.0f, py = 0.0f, pz = 0.0f;
        if (inR) {
            const float* pp = pos + (size_t)pidx * 3;
            px = pp[0]; py = pp[1]; pz = pp[2];
        }

        // B-matrix: low lanes carry (x,y), high lanes carry (z,1) of point m.
        v2f b;
        b.x = top ? px : pz;
        b.y = top ? py : 1.0f;

        v8f c = {};
        // 8 args: (neg_a, A, neg_b, B, c_mod, C, reuse_a, reuse_b)
        v8f d = __builtin_amdgcn_wmma_f32_16x16x4_f32(
            false, a, false, b, (short)0, c, false, false);

        // Lanes 0..15: d[0]=X', d[1]=Y', d[2]=Z' of point (base + lane).
        const float X = d[0];
        const float Y = d[1];
        const float Z = d[2];

        const float u = fx * X / Z + cx;
        const float v = fy * Y / Z + cy;
        const int xi = (int)u;   // v_cvt_i32_f32 truncates toward zero,
        const int yi = (int)v;   // matching jnp .astype(int32)

        if (top && inR && xi >= 0 && xi < W && yi >= 0 && yi < H) {
            atomicMax(&winner[yi * W + xi], pidx);   // no-return GLOBAL_ATOMIC_MAX_I32
        }
    }
}

// ---------------------------------------------------------------------------
// Kernel 2: resolve winners -> planar (1,3,H,W) image, clipped to [0,1].
// out index for channel ch at flat pixel p is simply ch*HW + p.
// ---------------------------------------------------------------------------
__global__ void gs_resolve(const int*   __restrict__ winner,
                           const float* __restrict__ colors,  // n x 3
                           float* __restrict__ out,
                           int npix) {
    int p = blockIdx.x * blockDim.x + threadIdx.x;
    if (p >= npix) return;
    int w = winner[p];
    float r = 0.0f, g = 0.0f, b = 0.0f;
    if (w >= 0) {
        const float* cc = colors + (size_t)w * 3;
        r = fminf(fmaxf(cc[0], 0.0f), 1.0f);
        g = fminf(fmaxf(cc[1], 0.0f), 1.0f);
        b = fminf(fmaxf(cc[2], 0.0f), 1.0f);
    }
    out[p]            = r;
    out[npix + p]     = g;
    out[2 * npix + p] = b;
}

extern "C" void kernel_launch(void* const* d_in, const int* in_sizes, int n_in,
                              void* d_out, int out_size, void* d_ws, size_t ws_size,
                              hipStream_t stream) {
    const float* pos  = (const float*)d_in[0];   // (N,3) f32
    const float* col  = (const float*)d_in[1];   // (N,3) f32
    const float* cam  = (const float*)d_in[2];   // (4,4) f32
    const float* intr = (const float*)d_in[3];   // (3,3) f32
    const int*   Hp   = (const int*)d_in[4];     // scalar
    const int*   Wp   = (const int*)d_in[5];     // scalar

    const int n    = in_sizes[0] / 3;            // number of points
    const int npix = out_size / 3;               // H*W
    int* winner = (int*)d_ws;                    // npix ints of scratch

    // 1) winner[p] = -1
    gs_fill_winner<<<(npix + 255) / 256, 256, 0, stream>>>(winner, npix);

    // 2) project + atomic-max scatter (grid-stride over 16-point wave chunks)
    const int pb = 256;                          // 8 waves / block
    int pgrid = (n + (pb / 32) * 16 - 1) / ((pb / 32) * 16);
    if (pgrid > 4096) pgrid = 4096;
    if (pgrid < 1)    pgrid = 1;
    gs_project_wmma<<<pgrid, pb, 0, stream>>>(pos, cam, intr, Hp, Wp, winner, n);

    // 3) gather colors -> planar image
    gs_resolve<<<(npix + 255) / 256, 256, 0, stream>>>(winner, col, (float*)d_out, npix);
}